// LossComposer_72739566125817
// MI455X (gfx1250) — compile-verified
//
#include <hip/hip_runtime.h>
#include <hip/hip_bf16.h>

namespace {
constexpr int NB  = 512;   // B
constexpr int NK  = 5;     // K
constexpr int NT  = 2048;  // T
constexpr int NP  = 2048;  // P
constexpr int NBK = NB * NK;
constexpr float TEMP = 0.8f;
}

typedef __attribute__((ext_vector_type(2))) float v2f;
typedef __attribute__((ext_vector_type(8))) float v8f;

#if defined(__HIP_DEVICE_COMPILE__) && __has_builtin(__builtin_amdgcn_wmma_f32_16x16x4_f32)
#define CDNA5_USE_WMMA 1
#else
#define CDNA5_USE_WMMA 0
#endif

__device__ __forceinline__ float waveSum(float v) {
  for (int off = 16; off > 0; off >>= 1) v += __shfl_down(v, off, 32);
  return v;
}

// ---------------------------------------------------------------------------
// Kernel 1: one workgroup per (b,k). Stages P/Q/m rows in LDS, computes
// masked sums, L1, bond loss, and H (3x3 cross-covariance) via WMMA.
// Thread 0 finishes the Kabsch closed form (Cardano eigenvalues of H^T H).
// ---------------------------------------------------------------------------
__global__ __launch_bounds__(256) void kabsch_local_kernel(
    const float* __restrict__ preds, const float* __restrict__ target,
    const int* __restrict__ mask,
    float* __restrict__ rmsd_out, float* __restrict__ local_out,
    float* __restrict__ l1_out) {
  __shared__ float ldsP[NT * 3];
  __shared__ float ldsQ[NT * 3];
  __shared__ float ldsM[NT];
  __shared__ float sAcc[21];  // [0..11] scalars, [12..20] H

  const int bk  = blockIdx.x;
  const int b   = bk / NK;
  const int tid = threadIdx.x;
  if (tid < 21) sAcc[tid] = 0.f;

  float nm = 0, sp0 = 0, sp1 = 0, sp2 = 0, sq0 = 0, sq1 = 0, sq2 = 0;
  float spp = 0, sqq = 0, sl1 = 0;
#if !CDNA5_USE_WMMA
  float hr[9] = {0, 0, 0, 0, 0, 0, 0, 0, 0};
#endif

  const float* Pbase = preds  + (size_t)bk * NT * 3;
  const float* Qbase = target + (size_t)b  * NT * 3;
  const int*   Mbase = mask   + (size_t)b  * NT;

  for (int t = tid; t < NT; t += 256) {
    float px = Pbase[t * 3 + 0], py = Pbase[t * 3 + 1], pz = Pbase[t * 3 + 2];
    float qx = Qbase[t * 3 + 0], qy = Qbase[t * 3 + 1], qz = Qbase[t * 3 + 2];
    float m  = (float)Mbase[t];
    ldsP[t * 3 + 0] = px; ldsP[t * 3 + 1] = py; ldsP[t * 3 + 2] = pz;
    ldsQ[t * 3 + 0] = qx; ldsQ[t * 3 + 1] = qy; ldsQ[t * 3 + 2] = qz;
    ldsM[t] = m;
    nm  += m;
    sp0 += m * px; sp1 += m * py; sp2 += m * pz;
    sq0 += m * qx; sq1 += m * qy; sq2 += m * qz;
    spp += m * (px * px + py * py + pz * pz);
    sqq += m * (qx * qx + qy * qy + qz * qz);
    sl1 += m * (fabsf(px - qx) + fabsf(py - qy) + fabsf(pz - qz));
#if !CDNA5_USE_WMMA
    hr[0] += m * px * qx; hr[1] += m * px * qy; hr[2] += m * px * qz;
    hr[3] += m * py * qx; hr[4] += m * py * qy; hr[5] += m * py * qz;
    hr[6] += m * pz * qx; hr[7] += m * pz * qy; hr[8] += m * pz * qz;
#endif
  }
  __syncthreads();

  // Local bond loss from LDS (neighbor access, no global re-read).
  float bnum = 0, bden = 0;
  for (int t = tid; t < NT - 1; t += 256) {
    float m2 = ldsM[t] * ldsM[t + 1];
    float dx = ldsP[(t + 1) * 3 + 0] - ldsP[t * 3 + 0];
    float dy = ldsP[(t + 1) * 3 + 1] - ldsP[t * 3 + 1];
    float dz = ldsP[(t + 1) * 3 + 2] - ldsP[t * 3 + 2];
    float pd = fminf(sqrtf(dx * dx + dy * dy + dz * dz + 1e-12f), 1e4f);
    float ex = ldsQ[(t + 1) * 3 + 0] - ldsQ[t * 3 + 0];
    float ey = ldsQ[(t + 1) * 3 + 1] - ldsQ[t * 3 + 1];
    float ez = ldsQ[(t + 1) * 3 + 2] - ldsQ[t * 3 + 2];
    float td = fminf(sqrtf(ex * ex + ey * ey + ez * ez + 1e-12f), 1e4f);
    float d  = pd - td;
    bnum += m2 * d * d;
    bden += m2;
  }

  {
    float acc[12] = {nm, sp0, sp1, sp2, sq0, sq1, sq2, spp, sqq, sl1, bnum, bden};
    const int lane = tid & 31;
    for (int i = 0; i < 12; ++i) {
      float v = waveSum(acc[i]);
      if (lane == 0) atomicAdd(&sAcc[i], v);
    }
#if !CDNA5_USE_WMMA
    for (int i = 0; i < 9; ++i) {
      float v = waveSum(hr[i]);
      if (lane == 0) atomicAdd(&sAcc[12 + i], v);
    }
#endif
  }

#if CDNA5_USE_WMMA
  // H = sum_t (m*P_t) outer (m*Q_t) as a 3xT * Tx3 GEMM on the WMMA unit.
  // A (16x4 f32): M=coordinate (rows 3..15 zeroed via sel), K=4 consecutive
  // t's. Layout: VGPR0 -> lanes 0-15: K=0, lanes 16-31: K=2; VGPR1 -> K=1/K=3.
  // B (4x16): N=coordinate, same K split across lane halves.
  // C (16x16 f32, 8 VGPRs): H[i][j] lands in VGPR i, lane j (i,j in 0..2).
  //
  // Branchless operand build: clamp the row index so the LDS address is
  // always in-bounds (one v_cndmask), and fold the zero-padding of rows
  // 3..15 into the mask multiply. All ds_loads are unconditional -> no
  // exec-mask diamonds, one dscnt wait per chunk, EXEC stays all-ones.
  {
    const int   lane = tid & 31;
    const int   wv   = tid >> 5;
    const int   half = lane >> 4;
    const int   r    = lane & 15;
    const int   rc   = (r < 3) ? r : 0;       // safe LDS row index
    const float sel  = (r < 3) ? 1.f : 0.f;   // zero-pad rows 3..15
    v8f c = {};
    const int t_begin = wv * (NT / 8);
    const int t_end   = t_begin + (NT / 8);
    for (int t0 = t_begin; t0 < t_end; t0 += 4) {
      const int   ta = t0 + half * 2;
      const float m0 = ldsM[ta] * sel;
      const float m1 = ldsM[ta + 1] * sel;
      v2f a, bb;
      a.x  = ldsP[ta * 3 + rc]       * m0;
      a.y  = ldsP[(ta + 1) * 3 + rc] * m1;
      bb.x = ldsQ[ta * 3 + rc]       * m0;
      bb.y = ldsQ[(ta + 1) * 3 + rc] * m1;
      c = __builtin_amdgcn_wmma_f32_16x16x4_f32(false, a, false, bb,
                                                (short)0, c, false, false);
    }
    if (half == 0 && r < 3) {
      atomicAdd(&sAcc[12 + 0 * 3 + r], c[0]);
      atomicAdd(&sAcc[12 + 1 * 3 + r], c[1]);
      atomicAdd(&sAcc[12 + 2 * 3 + r], c[2]);
    }
  }
#endif
  __syncthreads();

  if (tid == 0) {
    const float n    = sAcc[0];
    const float msum = fmaxf(n, 1e-8f);
    const float inv  = 1.f / msum;
    float sp[3] = {sAcc[1], sAcc[2], sAcc[3]};
    float sq[3] = {sAcc[4], sAcc[5], sAcc[6]};
    float h[9];
    for (int i = 0; i < 3; ++i)
      for (int j = 0; j < 3; ++j)
        h[i * 3 + j] = sAcc[12 + i * 3 + j] - sp[i] * sq[j] * inv;
    float sppc = sAcc[7] - (sp[0] * sp[0] + sp[1] * sp[1] + sp[2] * sp[2]) * inv;
    float sqqc = sAcc[8] - (sq[0] * sq[0] + sq[1] * sq[1] + sq[2] * sq[2]) * inv;

    // A = H^T H (symmetric), eigenvalues via Cardano.
    float a00 = h[0] * h[0] + h[3] * h[3] + h[6] * h[6];
    float a11 = h[1] * h[1] + h[4] * h[4] + h[7] * h[7];
    float a22 = h[2] * h[2] + h[5] * h[5] + h[8] * h[8];
    float a01 = h[0] * h[1] + h[3] * h[4] + h[6] * h[7];
    float a02 = h[0] * h[2] + h[3] * h[5] + h[6] * h[8];
    float a12 = h[1] * h[2] + h[4] * h[5] + h[7] * h[8];
    float q  = (a00 + a11 + a22) * (1.f / 3.f);
    float p1 = a01 * a01 + a02 * a02 + a12 * a12;
    float b00 = a00 - q, b11 = a11 - q, b22 = a22 - q;
    float p2 = b00 * b00 + b11 * b11 + b22 * b22 + 2.f * p1;
    float p  = sqrtf(fmaxf(p2, 0.f) / 6.f);
    float e0, e1, e2;
    if (p > 1e-30f) {
      float ip  = 1.f / p;
      float c00 = b00 * ip, c11 = b11 * ip, c22 = b22 * ip;
      float c01 = a01 * ip, c02 = a02 * ip, c12 = a12 * ip;
      float detB = c00 * (c11 * c22 - c12 * c12)
                 - c01 * (c01 * c22 - c12 * c02)
                 + c02 * (c01 * c12 - c11 * c02);
      float rr  = fminf(fmaxf(detB * 0.5f, -1.f), 1.f);
      float phi = acosf(rr) * (1.f / 3.f);
      e0 = q + 2.f * p * cosf(phi);
      e2 = q + 2.f * p * cosf(phi + 2.0943951023931953f);  // + 2*pi/3
      e1 = 3.f * q - e0 - e2;
    } else {
      e0 = e1 = e2 = q;
    }
    float s0 = sqrtf(fmaxf(e0, 0.f));
    float s1 = sqrtf(fmaxf(e1, 0.f));
    float s2 = sqrtf(fmaxf(e2, 0.f));
    float detH = h[0] * (h[4] * h[8] - h[5] * h[7])
               - h[1] * (h[3] * h[8] - h[5] * h[6])
               + h[2] * (h[3] * h[7] - h[4] * h[6]);
    float trRH   = s0 + s1 + (detH < 0.f ? -s2 : s2);
    float sum_sq = fmaxf(sppc + sqqc - 2.f * trRH, 0.f);
    float nvalid = fmaxf(3.f * n, 1.f);
    rmsd_out[bk]  = sqrtf(sum_sq / nvalid + 1e-8f);
    local_out[bk] = sAcc[10] / fmaxf(sAcc[11], 1.f);
    l1_out[bk]    = sAcc[9] / nvalid;
  }
}

// ---------------------------------------------------------------------------
// Kernel 2: pairwise distance loss, one workgroup per (b,k).
// ---------------------------------------------------------------------------
__global__ __launch_bounds__(256) void pairwise_kernel(
    const float* __restrict__ preds, const float* __restrict__ target,
    const int* __restrict__ i_idx, const int* __restrict__ j_idx,
    float* __restrict__ dist_out) {
  __shared__ float red[8];
  const int bk  = blockIdx.x;
  const int b   = bk / NK;
  const int tid = threadIdx.x;
  const float* Pb = preds  + (size_t)bk * NT * 3;
  const float* Qb = target + (size_t)b  * NT * 3;

  float acc = 0.f;
  for (int pp = tid; pp < NP; pp += 256) {
    int ii = i_idx[(size_t)b * NP + pp];
    int jj = j_idx[(size_t)b * NP + pp];
    ii = ii < 0 ? 0 : (ii > NT - 1 ? NT - 1 : ii);  // match JAX OOB clamping
    jj = jj < 0 ? 0 : (jj > NT - 1 ? NT - 1 : jj);
    float tx = Qb[ii * 3 + 0] - Qb[jj * 3 + 0];
    float ty = Qb[ii * 3 + 1] - Qb[jj * 3 + 1];
    float tz = Qb[ii * 3 + 2] - Qb[jj * 3 + 2];
    float td = sqrtf(tx * tx + ty * ty + tz * tz + 1e-12f);
    float px = Pb[ii * 3 + 0] - Pb[jj * 3 + 0];
    float py = Pb[ii * 3 + 1] - Pb[jj * 3 + 1];
    float pz = Pb[ii * 3 + 2] - Pb[jj * 3 + 2];
    float pd = sqrtf(px * px + py * py + pz * pz + 1e-12f);
    float d  = pd - td;
    acc += d * d;
  }
  float v = waveSum(acc);
  if ((tid & 31) == 0) red[tid >> 5] = v;
  __syncthreads();
  if (tid == 0) {
    float s = 0.f;
    for (int w = 0; w < 8; ++w) s += red[w];
    dist_out[bk] = s / (float)NP;
  }
}

// ---------------------------------------------------------------------------
// Kernel 3: softmin over K, mean over B, epoch-dependent combination.
// ---------------------------------------------------------------------------
__device__ __forceinline__ float softmin5(const float* l) {
  float mn = l[0];
  for (int k = 1; k < NK; ++k) mn = fminf(mn, l[k]);
  float s = 0.f, sw = 0.f;
  for (int k = 0; k < NK; ++k) {
    float w = expf((mn - l[k]) / TEMP);
    s  += w * l[k];
    sw += w;
  }
  return s / sw;
}

__global__ __launch_bounds__(512) void combine_kernel(
    const float* __restrict__ rmsd, const float* __restrict__ dist,
    const float* __restrict__ local, const float* __restrict__ l1,
    const int* __restrict__ epoch_p, float* __restrict__ out) {
  __shared__ float red[16 * 4];
  const int b = threadIdx.x;  // 512 threads == NB
  float v[4] = {0.f, 0.f, 0.f, 0.f};
  if (b < NB) {
    v[0] = softmin5(rmsd  + b * NK);
    v[1] = softmin5(dist  + b * NK);
    v[2] = softmin5(local + b * NK);
    v[3] = softmin5(l1    + b * NK);
  }
  const int wv = threadIdx.x >> 5, lane = threadIdx.x & 31;
  for (int i = 0; i < 4; ++i) {
    float s = waveSum(v[i]);
    if (lane == 0) red[wv * 4 + i] = s;
  }
  __syncthreads();
  if (threadIdx.x == 0) {
    float tot[4] = {0.f, 0.f, 0.f, 0.f};
    for (int w = 0; w < 16; ++w)
      for (int i = 0; i < 4; ++i) tot[i] += red[w * 4 + i];
    for (int i = 0; i < 4; ++i) tot[i] *= (1.f / (float)NB);
    const int ep = *epoch_p;
    float result;
    if (ep < 5) {                       // WARMUP
      result = tot[3];
    } else {
      float t    = (float)(ep - 5);
      float ramp = fminf(1.f, t * (1.f / 5.f));  // RAMP_EPOCHS
      result = tot[0] + 0.07f * ramp * tot[1] + 0.03f * tot[2];
    }
    out[0] = result;
  }
}

// ---------------------------------------------------------------------------
extern "C" void kernel_launch(void* const* d_in, const int* in_sizes, int n_in,
                              void* d_out, int out_size, void* d_ws, size_t ws_size,
                              hipStream_t stream) {
  (void)in_sizes; (void)n_in; (void)out_size; (void)ws_size;
  const float* preds   = (const float*)d_in[0];
  const float* target  = (const float*)d_in[1];
  const int*   mask    = (const int*)d_in[2];
  const int*   i_idx   = (const int*)d_in[3];
  const int*   j_idx   = (const int*)d_in[4];
  const int*   epoch_p = (const int*)d_in[5];

  float* ws    = (float*)d_ws;
  float* rmsd  = ws;
  float* local = ws + NBK;
  float* dist  = ws + 2 * NBK;
  float* l1    = ws + 3 * NBK;

  kabsch_local_kernel<<<NBK, 256, 0, stream>>>(preds, target, mask, rmsd, local, l1);
  pairwise_kernel<<<NBK, 256, 0, stream>>>(preds, target, i_idx, j_idx, dist);
  combine_kernel<<<1, 512, 0, stream>>>(rmsd, dist, local, l1, epoch_p, (float*)d_out);
}